// KNNNet_17016660427268
// MI455X (gfx1250) — compile-verified
//
#include <hip/hip_runtime.h>
#include <hip/hip_bf16.h>
#include <math.h>

typedef __attribute__((ext_vector_type(2))) float v2f;
typedef __attribute__((ext_vector_type(8))) float v8f;

#define NQ   16
#define NN   2048
#define DIM  256
#define KTOP 10

// ---------------------------------------------------------------------------
// Kernel 1: scores[b][j] = 2*q_b.n_j - ||q_b||^2 - ||n_j||^2  via f32 WMMA.
// One wave (32 threads) per 16-neighbor tile; 64 x V_WMMA_F32_16X16X4_F32.
// Norms are accumulated from the same fragment loads (every element of the
// 16 query rows / 16 neighbor rows passes through exactly one lane-half).
// ---------------------------------------------------------------------------
__global__ __launch_bounds__(32) void knn_scores_wmma(
    const float* __restrict__ q,    // [16][256]
    const float* __restrict__ nb,   // [2048][256]
    float* __restrict__ scores)     // [16][2048]
{
    const int lane  = threadIdx.x;      // 0..31
    const int col   = lane & 15;        // A row (M) and B column (N)
    const int half  = lane >> 4;        // 0: K={0,1}, 1: K={2,3} within each step
    const int nbase = blockIdx.x * 16;

    const float* __restrict__ qrow = q  + col * DIM;
    const float* __restrict__ nrow = nb + (size_t)(nbase + col) * DIM;

    v8f  acc   = {};
    float qpart = 0.0f;   // partial ||q_col||^2 (this half's K elements)
    float npart = 0.0f;   // partial ||n_{nbase+col}||^2

#pragma unroll 8
    for (int k = 0; k < DIM; k += 4) {
        v2f a, b;
        const int kk = k + 2 * half;
        a.x = qrow[kk + 0];
        a.y = qrow[kk + 1];
        b.x = nrow[kk + 0];
        b.y = nrow[kk + 1];
        qpart += a.x * a.x + a.y * a.y;
        npart += b.x * b.x + b.y * b.y;
        // D = A(16x4 f32) * B(4x16 f32) + C   -> v_wmma_f32_16x16x4_f32
        acc = __builtin_amdgcn_wmma_f32_16x16x4_f32(
            /*neg_a=*/false, a, /*neg_b=*/false, b,
            /*c_mod=*/(short)0, acc, /*reuse_a=*/false, /*reuse_b=*/false);
    }

    // Combine the two K-halves: lane L and lane L^16 hold the two halves of
    // the same row/column norm.
    qpart += __shfl_xor(qpart, 16, 32);
    npart += __shfl_xor(npart, 16, 32);
    // Now every lane holds ||q_{lane&15}||^2 in qpart and ||n_{nbase+(lane&15)}||^2
    // in npart.

    // C/D layout: VGPR v -> element (M = v + 8*half, N = col).
#pragma unroll
    for (int v = 0; v < 8; ++v) {
        const int m   = v + 8 * half;
        const float qn2 = __shfl(qpart, m, 32);   // lane m holds ||q_m||^2
        const float sc  = 2.0f * acc[v] - qn2 - npart;
        scores[(size_t)m * NN + nbase + col] = sc;
    }
}

// ---------------------------------------------------------------------------
// Kernel 2: Bsum[b][j] = sum_j' |s[b][j] - s[b][j']|
// grid = 16 batches * 8 slices, 256 threads; score row staged in LDS.
// ---------------------------------------------------------------------------
__global__ __launch_bounds__(256) void knn_bsum(
    const float* __restrict__ scores,  // [16][2048]
    float* __restrict__ bsum)          // [16][2048]
{
    const int b     = blockIdx.x >> 3;
    const int slice = blockIdx.x & 7;

    __shared__ float s[NN];
    const float* __restrict__ srow = scores + (size_t)b * NN;
    for (int t = threadIdx.x; t < NN; t += 256) s[t] = srow[t];
    __syncthreads();

    const int   j  = slice * 256 + threadIdx.x;
    const float sj = s[j];
    float sum = 0.0f;
#pragma unroll 8
    for (int jp = 0; jp < NN; ++jp) sum += fabsf(sj - s[jp]);
    bsum[(size_t)b * NN + j] = sum;
}

// ---------------------------------------------------------------------------
// Kernel 3: per batch, for rows i = 0..K-1 find
//   argmax_j ( (N-1-2i)*s_j - Bsum_j )   (ties -> lowest j, like jnp.argmax)
// and emit the summed one-hot row.
// ---------------------------------------------------------------------------
__global__ __launch_bounds__(256) void knn_topk(
    const float* __restrict__ scores,
    const float* __restrict__ bsum,
    float* __restrict__ out)           // [16][2048]
{
    const int b   = blockIdx.x;
    const int tid = threadIdx.x;

    __shared__ float bestv[KTOP][256];
    __shared__ int   besti[KTOP][256];

    const float* __restrict__ srow = scores + (size_t)b * NN;
    const float* __restrict__ brow = bsum   + (size_t)b * NN;
    float* __restrict__ orow = out + (size_t)b * NN;

    float bv[KTOP];
    int   bi[KTOP];
#pragma unroll
    for (int i = 0; i < KTOP; ++i) { bv[i] = -INFINITY; bi[i] = NN; }

    for (int j = tid; j < NN; j += 256) {
        const float sj = srow[j];
        const float Bj = brow[j];
        orow[j] = 0.0f;                 // fully overwrite the output row
#pragma unroll
        for (int i = 0; i < KTOP; ++i) {
            const float c   = (float)(NN - 1 - 2 * i);
            const float val = c * sj - Bj;
            if (val > bv[i]) { bv[i] = val; bi[i] = j; }
            // ascending-j loop within a thread => first max kept automatically
        }
    }

#pragma unroll
    for (int i = 0; i < KTOP; ++i) { bestv[i][tid] = bv[i]; besti[i][tid] = bi[i]; }
    __syncthreads();

    for (int stride = 128; stride > 0; stride >>= 1) {
        if (tid < stride) {
#pragma unroll
            for (int i = 0; i < KTOP; ++i) {
                const float v2 = bestv[i][tid + stride];
                const int   i2 = besti[i][tid + stride];
                const float v1 = bestv[i][tid];
                const int   i1 = besti[i][tid];
                if (v2 > v1 || (v2 == v1 && i2 < i1)) {
                    bestv[i][tid] = v2; besti[i][tid] = i2;
                }
            }
        }
        __syncthreads();
    }

    if (tid == 0) {
#pragma unroll
        for (int i = 0; i < KTOP; ++i) {
            orow[besti[i][0]] += 1.0f;  // duplicates accumulate, matching ref
        }
    }
}

// ---------------------------------------------------------------------------
extern "C" void kernel_launch(void* const* d_in, const int* in_sizes, int n_in,
                              void* d_out, int out_size, void* d_ws, size_t ws_size,
                              hipStream_t stream) {
    (void)in_sizes; (void)n_in; (void)out_size; (void)ws_size;
    const float* q  = (const float*)d_in[0];   // [16][256]
    const float* nb = (const float*)d_in[1];   // [2048][256]
    float* out      = (float*)d_out;           // [16][2048]

    float* scores = (float*)d_ws;              // 16*2048 f32 = 128 KB
    float* bsum   = scores + NQ * NN;          // another 128 KB

    knn_scores_wmma<<<NN / 16, 32, 0, stream>>>(q, nb, scores);
    knn_bsum<<<NQ * 8, 256, 0, stream>>>(scores, bsum);
    knn_topk<<<NQ, 256, 0, stream>>>(scores, bsum, out);
}